// AdaptiveThresholdCoding_50929722196682
// MI455X (gfx1250) — compile-verified
//
#include <hip/hip_runtime.h>

// Adaptive-threshold spike encoding for MI455X (gfx1250, wave32).
//
// x:   [B=32, 256, 1024] float32  (N = 8,388,608 elements, 32 MiB)
// out: [B=32, T=32, 256, 1024] float32 (1 GiB)  out[b][t][f]
//
// Memory-bound: ~1.09 GB total traffic -> ~47 us roofline at 23.3 TB/s.
// Strategy: one thread per float4; 32-step recurrence fully unrolled in
// registers; 1 nontemporal b128 load + 32 nontemporal b128 stores per thread.

typedef __attribute__((ext_vector_type(4))) float v4f;

#define T_STEPS 32
#define FEAT    (256 * 1024)   // per-batch feature count = 262144
#define FVEC    (FEAT / 4)     // 65536 float4 per batch slice
#define RATE    0.1f
#define ONE_M_R 0.9f
#define THR0    0.5f

__global__ __launch_bounds__(256) void
AdaptiveThresholdCoding_spike_kernel(const v4f* __restrict__ x,
                                     v4f* __restrict__ out)
{
    // Global vec4 index over the whole input. 2,097,152 total -> fits u32.
    const unsigned i  = blockIdx.x * 256u + threadIdx.x;
    const unsigned b  = i >> 16;        // FVEC == 65536
    const unsigned f4 = i & 0xFFFFu;

    // Input is read exactly once per element: NT load, b128.
    const v4f xv = __builtin_nontemporal_load(&x[i]);

    v4f acc = {0.0f, 0.0f, 0.0f, 0.0f};
    v4f thr = {THR0, THR0, THR0, THR0};
    v4f rax;
#pragma unroll
    for (int c = 0; c < 4; ++c)
        rax[c] = RATE * __builtin_fabsf(xv[c]);

    // out[b][t][:] base; 64-bit addressing (output spans 1 GiB).
    v4f* __restrict__ outp = out + (size_t)b * (T_STEPS * (size_t)FVEC) + f4;

#pragma unroll
    for (int t = 0; t < T_STEPS; ++t) {
        v4f sp;
#pragma unroll
        for (int c = 0; c < 4; ++c) {
            const float a    = acc[c] + xv[c];
            const bool  fire = a >= thr[c];
            sp[c]  = fire ? 1.0f : 0.0f;
            acc[c] = fire ? 0.0f : a;
            // thr = thr*(1-rate) + rate*|x|
            thr[c] = __builtin_fmaf(thr[c], ONE_M_R, rax[c]);
        }
        // Write-once 1 GiB stream: NT store (th:TH_STORE_NT) so it does not
        // thrash the 192 MB L2. 32 lanes x 16 B = 512 B contiguous per wave.
        __builtin_nontemporal_store(sp, outp + (size_t)t * FVEC);
    }
}

extern "C" void kernel_launch(void* const* d_in, const int* in_sizes, int n_in,
                              void* d_out, int out_size, void* d_ws, size_t ws_size,
                              hipStream_t stream)
{
    (void)n_in; (void)out_size; (void)d_ws; (void)ws_size;

    const v4f* x   = (const v4f*)d_in[0];
    v4f*       out = (v4f*)d_out;

    const int nvec   = in_sizes[0] / 4;        // 2,097,152
    const int blocks = (nvec + 255) / 256;     // 8192

    AdaptiveThresholdCoding_spike_kernel<<<blocks, 256, 0, stream>>>(x, out);
}